// PredictiveCodingColumn_38869454029214
// MI455X (gfx1250) — compile-verified
//
#include <hip/hip_runtime.h>
#include <stdint.h>

// MI455X / gfx1250 implementation.
// One wave32-lane per channel; all per-channel state lives in VGPRs.
// The dominant 192 MB noise stream is staged global->LDS with CDNA5 async
// loads: all 16 per-thread async loads (8 substeps x {pop,inh}) are issued
// up-front into 8 LDS buffers, then drained with s_wait_asynccnt using
// in-order async retirement — 16-deep per-wave MLP on the noise stream while
// the Izhikevich VALU work runs. Everything is single-pass -> NT hints.

typedef float f4 __attribute__((ext_vector_type(4)));
typedef float f2 __attribute__((ext_vector_type(2)));

#define NCH  1000000
#define NPOP (NCH * 4)
#define NINH (NCH * 2)
#define TPB  256
#define NSUB 8

__device__ __forceinline__ float fclampf(float x, float lo, float hi) {
  return fminf(fmaxf(x, lo), hi);
}

// Izhikevich update, matching the reference order exactly.
__device__ __forceinline__ void izh(float& v, float& u, float& r, float I,
                                    float a, float b, float c, float d,
                                    float tonic) {
  I += tonic;
  v = v + (0.04f * v * v + 5.0f * v + 140.0f - u + I);
  u = u + a * (b * v - u);
  const bool sp = (v >= 30.0f);
  r = 0.9f * r + (sp ? 0.1f : 0.0f);
  v = sp ? c : v;
  u = sp ? (u + d) : u;
}

#define LD4(dst, src)                                                         \
  {                                                                           \
    f4 t_ = __builtin_nontemporal_load(&((const f4*)(src))[c4]);              \
    dst[0] = t_.x; dst[1] = t_.y; dst[2] = t_.z; dst[3] = t_.w;               \
  }
#define LD2(dst, src)                                                         \
  {                                                                           \
    f2 t_ = __builtin_nontemporal_load(&((const f2*)(src))[c4]);              \
    dst[0] = t_.x; dst[1] = t_.y;                                             \
  }

// Async global -> LDS (CDNA5, ASYNCcnt-tracked), non-temporal: noise is
// consumed exactly once. LDS dest offset in a 32-bit VGPR, global address in
// a 64-bit VGPR pair (GV mode).
#define ASYNC_B128(lds, gaddr)                                                \
  asm volatile("global_load_async_to_lds_b128 %0, %1, off th:TH_LOAD_NT"      \
               :: "v"(lds), "v"(gaddr) : "memory")
#define ASYNC_B64(lds, gaddr)                                                 \
  asm volatile("global_load_async_to_lds_b64 %0, %1, off th:TH_LOAD_NT"       \
               :: "v"(lds), "v"(gaddr) : "memory")
// In-order async retirement: waiting for <= n outstanding guarantees the
// first (16 - n) issued loads have landed in LDS.
#define WAIT_ASYNC(n)                                                         \
  asm volatile("s_wait_asynccnt %0" :: "n"(n) : "memory")

#define ST_NT(ptr, val) __builtin_nontemporal_store((val), (ptr))

__global__ __launch_bounds__(TPB)
void PredictiveCodingColumn_38869454029214_kernel(
    const float* __restrict__ sens,      const float* __restrict__ pred,
    const float* __restrict__ noise_pop, const float* __restrict__ noise_inh,
    const float* __restrict__ g_vup, const float* __restrict__ g_uup, const float* __restrict__ g_rup,
    const float* __restrict__ g_vdn, const float* __restrict__ g_udn, const float* __restrict__ g_rdn,
    const float* __restrict__ g_vdp, const float* __restrict__ g_udp, const float* __restrict__ g_rdp,
    const float* __restrict__ g_vin, const float* __restrict__ g_uin, const float* __restrict__ g_rin,
    const float* __restrict__ g_pbias, const float* __restrict__ g_elig,
    float* __restrict__ out, float* __restrict__ block_fe) {
  __shared__ f4 smp[NSUB][TPB];  // 32 KB: per-lane noise_pop slots
  __shared__ f2 smi[NSUB][TPB];  // 16 KB: per-lane noise_inh slots
  __shared__ float red[TPB];     // free-energy block reduction

  const int tid = threadIdx.x;
  const int ch = blockIdx.x * TPB + tid;
  const bool active = (ch < NCH);
  const int c4 = active ? ch : (NCH - 1);  // clamp so barriers stay uniform

  // Issue the entire noise pipeline: 16 async loads per lane, in substep
  // order (pop[s], inh[s]).
#pragma unroll
  for (int s = 0; s < NSUB; ++s) {
    const uint32_t lp = (uint32_t)(uintptr_t)&smp[s][tid];
    const uint32_t li = (uint32_t)(uintptr_t)&smi[s][tid];
    const uint64_t gp =
        (uint64_t)(uintptr_t)(noise_pop + (size_t)s * NPOP + (size_t)c4 * 4);
    const uint64_t gi =
        (uint64_t)(uintptr_t)(noise_inh + (size_t)s * NINH + (size_t)c4 * 2);
    ASYNC_B128(lp, gp);
    ASYNC_B64(li, gi);
  }

  // Channel-constant drives (single-use -> NT).
  const float sd = __builtin_nontemporal_load(&sens[c4]);
  const float pb = __builtin_nontemporal_load(&g_pbias[c4]);
  const float I_sens = sd * 15.0f;                                       // G_SENS
  const float I_pred = (__builtin_nontemporal_load(&pred[c4]) + pb) * 12.0f;
  const float I_spdn = I_pred - sd * 5.0f;                               // -G_PREC*sens
  const float el_in = __builtin_nontemporal_load(&g_elig[c4]);

  // State (read once -> nontemporal b128/b64).
  float vu[4], uu[4], ru[4], vd[4], ud[4], rd[4], vp[4], up[4], rp[4];
  float vi[2], ui[2], ri[2];
  LD4(vu, g_vup); LD4(uu, g_uup); LD4(ru, g_rup);
  LD4(vd, g_vdn); LD4(ud, g_udn); LD4(rd, g_rdn);
  LD4(vp, g_vdp); LD4(up, g_udp); LD4(rp, g_rdp);
  LD2(vi, g_vin); LD2(ui, g_uin); LD2(ri, g_rin);

#pragma unroll
  for (int s = 0; s < NSUB; ++s) {
    // Need the first 2*(s+1) async loads complete -> allow 14-2s outstanding.
    WAIT_ASYNC(14 - 2 * s);

    const f4 np4 = smp[s][tid] * 0.4f;  // NOISE_SCALE
    const f2 ni2 = smi[s][tid] * 0.4f;
    const float np[4] = {np4.x, np4.y, np4.z, np4.w};
    const float ni[2] = {ni2.x, ni2.y};

    // I_asc uses previous-substep SP rates: mean(ru)*8 - mean(rd)*4.
    const float I_asc = (ru[0] + ru[1] + ru[2] + ru[3]) * (0.25f * 8.0f)
                      - (rd[0] + rd[1] + rd[2] + rd[3]) * (0.25f * 4.0f);
#pragma unroll
    for (int j = 0; j < 4; ++j)  // deep pyramidal (IB)
      izh(vp[j], up[j], rp[j], I_pred + I_asc + np[j],
          0.02f, 0.2f, -55.0f, 4.0f, 0.0f);

    const float I_in = (rp[0] + rp[1] + rp[2] + rp[3]) * (0.25f * 12.0f);
#pragma unroll
    for (int j = 0; j < 2; ++j)  // inhibitory (FS)
      izh(vi[j], ui[j], ri[j], I_in + ni[j],
          0.10f, 0.2f, -65.0f, 2.0f, -0.5f);

    const float inh5 = (ri[0] + ri[1]) * (0.5f * 5.0f);  // inh_ch * G_PREC
#pragma unroll
    for (int j = 0; j < 4; ++j)  // SP+ (RS)
      izh(vu[j], uu[j], ru[j], I_sens - inh5 + np[j],
          0.02f, 0.2f, -65.0f, 8.0f, -1.0f);
#pragma unroll
    for (int j = 0; j < 4; ++j)  // SP- (RS)
      izh(vd[j], ud[j], rd[j], I_spdn + np[j],
          0.02f, 0.2f, -65.0f, 8.0f, -1.0f);
  }

  // Channel-level outputs.
  const float sp_up = (ru[0] + ru[1] + ru[2] + ru[3]) * 0.25f;
  const float sp_dn = (rd[0] + rd[1] + rd[2] + rd[3]) * 0.25f;
  const float dpch  = (rp[0] + rp[1] + rp[2] + rp[3]) * 0.25f;
  const float pe_s = sp_up - sp_dn;
  const float pe_m = sp_up + sp_dn;
  const float el = el_in * 0.9f + (sp_up * sd - sp_dn * dpch);
  const float pbn = fclampf(pb + 0.005f * el, -2.0f, 2.0f);
  const float pgrad = 0.5f / (5.0f + 1e-4f) - 0.5f * pe_s * pe_s;
  const float gpn = fclampf(5.0f + 0.001f * pgrad, 0.5f, 15.0f);

  // d_out layout (return order): pe_signed[1M], pe_magnitude[1M],
  // free_energy[1], g_prec_new[1M], dp_ch[1M], sp_up_ch[1M], sp_dn_ch[1M],
  // pred_bias_new[1M], elig[1M]. Write-once -> NT stores.
  if (active) {
    ST_NT(&out[(size_t)0 * NCH + ch],     pe_s);
    ST_NT(&out[(size_t)1 * NCH + ch],     pe_m);
    ST_NT(&out[(size_t)2 * NCH + 1 + ch], gpn);
    ST_NT(&out[(size_t)3 * NCH + 1 + ch], dpch);
    ST_NT(&out[(size_t)4 * NCH + 1 + ch], sp_up);
    ST_NT(&out[(size_t)5 * NCH + 1 + ch], sp_dn);
    ST_NT(&out[(size_t)6 * NCH + 1 + ch], pbn);
    ST_NT(&out[(size_t)7 * NCH + 1 + ch], el);
  }

  // Deterministic block partial of 0.5 * g_prec * pe_m^2 (g_prec == 5).
  red[tid] = active ? (2.5f * pe_m * pe_m) : 0.0f;
  __syncthreads();
#pragma unroll
  for (int off = TPB / 2; off > 0; off >>= 1) {
    if (tid < off) red[tid] += red[tid + off];
    __syncthreads();
  }
  if (tid == 0) block_fe[blockIdx.x] = red[0];
}

__global__ __launch_bounds__(TPB)
void PredictiveCodingColumn_fe_reduce(const float* __restrict__ partials,
                                      int n, float* __restrict__ fe_out) {
  __shared__ float red[TPB];
  float s = 0.0f;
  for (int i = threadIdx.x; i < n; i += TPB) s += partials[i];
  red[threadIdx.x] = s;
  __syncthreads();
#pragma unroll
  for (int off = TPB / 2; off > 0; off >>= 1) {
    if (threadIdx.x < off) red[threadIdx.x] += red[threadIdx.x + off];
    __syncthreads();
  }
  if (threadIdx.x == 0) fe_out[0] = red[0];
}

extern "C" void kernel_launch(void* const* d_in, const int* in_sizes, int n_in,
                              void* d_out, int out_size, void* d_ws,
                              size_t ws_size, hipStream_t stream) {
  (void)in_sizes; (void)n_in; (void)out_size; (void)ws_size;
  const float* sens      = (const float*)d_in[0];
  const float* pred      = (const float*)d_in[1];
  const float* noise_pop = (const float*)d_in[2];
  const float* noise_inh = (const float*)d_in[3];
  const float* vup = (const float*)d_in[4];
  const float* uup = (const float*)d_in[5];
  const float* rup = (const float*)d_in[6];
  const float* vdn = (const float*)d_in[7];
  const float* udn = (const float*)d_in[8];
  const float* rdn = (const float*)d_in[9];
  const float* vdp = (const float*)d_in[10];
  const float* udp = (const float*)d_in[11];
  const float* rdp = (const float*)d_in[12];
  const float* vin = (const float*)d_in[13];
  const float* uin = (const float*)d_in[14];
  const float* rin = (const float*)d_in[15];
  const float* pbias = (const float*)d_in[16];
  const float* elig  = (const float*)d_in[17];

  float* out = (float*)d_out;
  float* block_fe = (float*)d_ws;  // one partial per block

  const int nblocks = (NCH + TPB - 1) / TPB;  // 3907
  PredictiveCodingColumn_38869454029214_kernel<<<nblocks, TPB, 0, stream>>>(
      sens, pred, noise_pop, noise_inh,
      vup, uup, rup, vdn, udn, rdn, vdp, udp, rdp, vin, uin, rin,
      pbias, elig, out, block_fe);

  PredictiveCodingColumn_fe_reduce<<<1, TPB, 0, stream>>>(
      block_fe, nblocks, out + (size_t)2 * NCH);
}